// CrossAttention_86638080295520
// MI455X (gfx1250) — compile-verified
//
#include <hip/hip_runtime.h>
#include <math.h>

typedef __attribute__((ext_vector_type(2))) float v2f;
typedef __attribute__((ext_vector_type(8))) float v8f;

constexpr int C_  = 32;
constexpr int B_  = 8;
constexpr int HW  = 384 * 384;            // 147456
constexpr int PIX_PER_WAVE = 512;         // 128 wmma-iterations of 4 pixels
constexpr int WAVES_PER_BLOCK = 8;        // 256 threads (wave32)
constexpr int BLOCKS_PER_BATCH = HW / (PIX_PER_WAVE * WAVES_PER_BLOCK); // 36
constexpr int TILES_PER_WAVE = 32;        // 32 tiles * 16 px = 512 px per wave

// ---------------------------------------------------------------------------
// Zero-init scratch (gram + channel sums) — d_ws is poisoned by the harness.
// ---------------------------------------------------------------------------
__global__ void zero_kernel(float* __restrict__ p, int n) {
    int i = blockIdx.x * blockDim.x + threadIdx.x;
    if (i < n) p[i] = 0.0f;
}

// ---------------------------------------------------------------------------
// Pass 1: per-batch 32x32 gram G = q @ v^T over hw, plus channel sums sq, sv.
// Four V_WMMA_F32_16X16X4_F32 per 4-pixel step cover the 32x32 output.
// q is streamed non-temporally (read exactly once) so v stays L2-resident for
// pass 3 (v = 151 MB < 192 MB L2).
// ---------------------------------------------------------------------------
__global__ void gram_kernel(const float* __restrict__ q, const float* __restrict__ v,
                            float* __restrict__ G, float* __restrict__ sq,
                            float* __restrict__ sv) {
    const int wave = threadIdx.x >> 5;
    const int lane = threadIdx.x & 31;
    const int b    = blockIdx.x / BLOCKS_PER_BATCH;
    const int blk  = blockIdx.x % BLOCKS_PER_BATCH;
    const int n0   = (blk * WAVES_PER_BLOCK + wave) * PIX_PER_WAVE;

    const int r   = lane & 15;         // matrix row (channel) this lane feeds
    const int off = (lane >> 4) * 2;   // lanes 16-31 carry K=2,3 / pixels +2,+3

    const float* qrow0 = q + (size_t)b * C_ * HW + (size_t)r * HW;
    const float* qrow1 = qrow0 + (size_t)16 * HW;
    const float* vrow0 = v + (size_t)b * C_ * HW + (size_t)r * HW;
    const float* vrow1 = vrow0 + (size_t)16 * HW;

    v8f g00 = {}, g01 = {}, g10 = {}, g11 = {};
    float s_q0 = 0.f, s_q1 = 0.f, s_v0 = 0.f, s_v1 = 0.f;

    for (int n = n0; n < n0 + PIX_PER_WAVE; n += 4) {
        v2f aq0 = __builtin_nontemporal_load((const v2f*)(qrow0 + n + off)); // q rows 0..15  (A)
        v2f aq1 = __builtin_nontemporal_load((const v2f*)(qrow1 + n + off)); // q rows 16..31
        v2f bv0 = *(const v2f*)(vrow0 + n + off);   // v rows 0..15  (B) — keep in L2
        v2f bv1 = *(const v2f*)(vrow1 + n + off);   // v rows 16..31
        s_q0 += aq0.x + aq0.y;  s_q1 += aq1.x + aq1.y;
        s_v0 += bv0.x + bv0.y;  s_v1 += bv1.x + bv1.y;
        g00 = __builtin_amdgcn_wmma_f32_16x16x4_f32(false, aq0, false, bv0, (short)0, g00, false, false);
        g01 = __builtin_amdgcn_wmma_f32_16x16x4_f32(false, aq0, false, bv1, (short)0, g01, false, false);
        g10 = __builtin_amdgcn_wmma_f32_16x16x4_f32(false, aq1, false, bv0, (short)0, g10, false, false);
        g11 = __builtin_amdgcn_wmma_f32_16x16x4_f32(false, aq1, false, bv1, (short)0, g11, false, false);
    }

    // Channel-sum partials: lanes {r, r+16} both cover row r (disjoint pixels).
    atomicAdd(&sq[b * C_ + r],      s_q0);
    atomicAdd(&sq[b * C_ + r + 16], s_q1);
    atomicAdd(&sv[b * C_ + r],      s_v0);
    atomicAdd(&sv[b * C_ + r + 16], s_v1);

    // D layout: VGPR i -> row i (lanes 0-15) / row i+8 (lanes 16-31), col = lane&15.
    float* Gb = G + b * C_ * C_;
    const int drow = (lane >> 4) ? 8 : 0;
    const int col  = lane & 15;
#pragma unroll
    for (int i = 0; i < 8; ++i) {
        atomicAdd(&Gb[(i + drow)      * C_ + col],      g00[i]);
        atomicAdd(&Gb[(i + drow)      * C_ + col + 16], g01[i]);
        atomicAdd(&Gb[(i + drow + 16) * C_ + col],      g10[i]);
        atomicAdd(&Gb[(i + drow + 16) * C_ + col + 16], g11[i]);
    }
}

// ---------------------------------------------------------------------------
// Pass 2 (tiny): per batch build S_raw from G/sq/sv, softmax rows, fold the
// three 1x1-conv weight matrices into M = wo@S@wa + I and cvec = wo@S@ba + bo.
// ---------------------------------------------------------------------------
__global__ void prep_kernel(const float* __restrict__ G, const float* __restrict__ sq,
                            const float* __restrict__ sv,
                            const float* __restrict__ wa, const float* __restrict__ ba,
                            const float* __restrict__ wb, const float* __restrict__ bbias,
                            const float* __restrict__ wc, const float* __restrict__ bc,
                            const float* __restrict__ wo, const float* __restrict__ bo,
                            float* __restrict__ M, float* __restrict__ cvec) {
    __shared__ float sG[1024], sH[1024], sS[1024], sT[1024];
    __shared__ float su[32], sw[32], sp[32];
    const int b = blockIdx.x;
    const int t = threadIdx.x;

    for (int i = t; i < 1024; i += 256) sG[i] = G[b * 1024 + i];
    __syncthreads();

    // H[i,d] = sum_j G[i,j] * wb[d,j]
    for (int e = t; e < 1024; e += 256) {
        int i = e >> 5, d = e & 31;
        float acc = 0.f;
        for (int j = 0; j < 32; ++j) acc += sG[i * 32 + j] * wb[d * 32 + j];
        sH[e] = acc;
    }
    if (t < 32) {
        float a0 = 0.f, a1 = 0.f;
        for (int i = 0; i < 32; ++i) {
            a0 += wc[t * 32 + i] * sq[b * 32 + i];   // u[c] = (wc @ sq)[c]
            a1 += wb[t * 32 + i] * sv[b * 32 + i];   // w[d] = (wb @ sv)[d]
        }
        su[t] = a0; sw[t] = a1;
    }
    __syncthreads();

    // S_raw[c,d] = sum_i wc[c,i] H[i,d] + bc[c]*w[d] + bb[d]*u[c] + HW*bc[c]*bb[d]
    for (int e = t; e < 1024; e += 256) {
        int c = e >> 5, d = e & 31;
        float acc = 0.f;
        for (int i = 0; i < 32; ++i) acc += wc[c * 32 + i] * sH[i * 32 + d];
        sS[e] = acc + bc[c] * sw[d] + bbias[d] * su[c] + (float)HW * bc[c] * bbias[d];
    }
    __syncthreads();

    // Row softmax (one thread per row; trivial size)
    if (t < 32) {
        float m = -INFINITY;
        for (int d = 0; d < 32; ++d) m = fmaxf(m, sS[t * 32 + d]);
        float s = 0.f;
        for (int d = 0; d < 32; ++d) { float e = expf(sS[t * 32 + d] - m); sS[t * 32 + d] = e; s += e; }
        float inv = 1.0f / s;
        for (int d = 0; d < 32; ++d) sS[t * 32 + d] *= inv;
    }
    __syncthreads();

    // T = S @ wa ; p[c] = S @ ba
    for (int e = t; e < 1024; e += 256) {
        int c = e >> 5, j = e & 31;
        float acc = 0.f;
        for (int d = 0; d < 32; ++d) acc += sS[c * 32 + d] * wa[d * 32 + j];
        sT[e] = acc;
    }
    if (t < 32) {
        float acc = 0.f;
        for (int d = 0; d < 32; ++d) acc += sS[t * 32 + d] * ba[d];
        sp[t] = acc;
    }
    __syncthreads();

    // M = wo @ T + I ; cvec = wo @ p + bo
    for (int e = t; e < 1024; e += 256) {
        int o = e >> 5, j = e & 31;
        float acc = (o == j) ? 1.0f : 0.0f;
        for (int c = 0; c < 32; ++c) acc += wo[o * 32 + c] * sT[c * 32 + j];
        M[b * 1024 + e] = acc;
    }
    if (t < 32) {
        float acc = bo[t];
        for (int c = 0; c < 32; ++c) acc += wo[t * 32 + c] * sp[c];
        cvec[b * 32 + t] = acc;
    }
}

// ---------------------------------------------------------------------------
// Pass 3: out[b] = M[b] @ v[b] + cvec[b]. M operands preloaded once per wave,
// 16 WMMAs per 32ch x 16px output tile (8 K-steps x 2 row blocks).
// v reads should hit L2 (kept resident by pass 1); out stores are
// non-temporal (write-once, never re-read) to avoid evicting v.
// ---------------------------------------------------------------------------
__global__ void apply_kernel(const float* __restrict__ v, const float* __restrict__ M,
                             const float* __restrict__ cvec, float* __restrict__ out) {
    const int wave = threadIdx.x >> 5;
    const int lane = threadIdx.x & 31;
    const int b    = blockIdx.x / BLOCKS_PER_BATCH;
    const int blk  = blockIdx.x % BLOCKS_PER_BATCH;
    const int pix0 = (blk * WAVES_PER_BLOCK + wave) * (16 * TILES_PER_WAVE);

    const int r   = lane & 15;
    const int off = (lane >> 4) * 2;

    // Preload A operands (M tiles): 8 K-steps x 2 output-row blocks
    const float* Mb = M + b * 1024;
    v2f A0[8], A1[8];
#pragma unroll
    for (int k = 0; k < 8; ++k) {
        A0[k] = *(const v2f*)(Mb + r * 32 + 4 * k + off);
        A1[k] = *(const v2f*)(Mb + (r + 16) * 32 + 4 * k + off);
    }
    const int drow = (lane >> 4) ? 8 : 0;
    float cv0[8], cv1[8];
#pragma unroll
    for (int i = 0; i < 8; ++i) {
        cv0[i] = cvec[b * 32 + i + drow];
        cv1[i] = cvec[b * 32 + i + drow + 16];
    }

    const float* vb = v   + (size_t)b * C_ * HW;
    float*       ob = out + (size_t)b * C_ * HW;
    const int col = lane & 15;

    for (int tile = 0; tile < TILES_PER_WAVE; ++tile) {
        const int p = pix0 + tile * 16;
        v8f d0 = {}, d1 = {};
#pragma unroll
        for (int k = 0; k < 8; ++k) {
            const int ch = 4 * k + off;            // K rows this lane-half carries
            v2f bv;
            bv.x = vb[(size_t)ch * HW + p + r];
            bv.y = vb[(size_t)(ch + 1) * HW + p + r];
            d0 = __builtin_amdgcn_wmma_f32_16x16x4_f32(false, A0[k], false, bv, (short)0, d0, false, false);
            d1 = __builtin_amdgcn_wmma_f32_16x16x4_f32(false, A1[k], false, bv, (short)0, d1, false, false);
        }
#pragma unroll
        for (int i = 0; i < 8; ++i) {
            __builtin_nontemporal_store(d0[i] + cv0[i], &ob[(size_t)(i + drow)      * HW + p + col]);
            __builtin_nontemporal_store(d1[i] + cv1[i], &ob[(size_t)(i + drow + 16) * HW + p + col]);
        }
    }
}

// ---------------------------------------------------------------------------
extern "C" void kernel_launch(void* const* d_in, const int* in_sizes, int n_in,
                              void* d_out, int out_size, void* d_ws, size_t ws_size,
                              hipStream_t stream) {
    const float* q  = (const float*)d_in[0];
    const float* v  = (const float*)d_in[1];
    const float* wa = (const float*)d_in[2];
    const float* ba = (const float*)d_in[3];
    const float* wb = (const float*)d_in[4];
    const float* bb = (const float*)d_in[5];
    const float* wc = (const float*)d_in[6];
    const float* bc = (const float*)d_in[7];
    const float* wo = (const float*)d_in[8];
    const float* bo = (const float*)d_in[9];
    float* out = (float*)d_out;

    // Workspace layout (floats): G[8*1024] | sq[8*32] | sv[8*32] | M[8*1024] | cvec[8*32]
    float* G  = (float*)d_ws;
    float* sq = G  + B_ * 1024;
    float* sv = sq + B_ * 32;
    float* M  = sv + B_ * 32;
    float* cv = M  + B_ * 1024;

    const int nzero = B_ * 1024 + 2 * B_ * 32;   // G + sq + sv
    zero_kernel<<<(nzero + 255) / 256, 256, 0, stream>>>(G, nzero);

    gram_kernel<<<B_ * BLOCKS_PER_BATCH, 256, 0, stream>>>(q, v, G, sq, sv);

    prep_kernel<<<B_, 256, 0, stream>>>(G, sq, sv, wa, ba, wb, bb, wc, bc, wo, bo, M, cv);

    apply_kernel<<<B_ * BLOCKS_PER_BATCH, 256, 0, stream>>>(v, M, cv, out);
}